// BaseGCN_19997367730676
// MI455X (gfx1250) — compile-verified
//
#include <hip/hip_runtime.h>
#include <hip/hip_bf16.h>

// ---------------------------------------------------------------------------
// 20-layer GCN for MI455X (gfx1250, wave32).
//  - degrees computed once per launch (self-loop folded in as init=1.0)
//  - self-loop edge handled by initializing agg := hw in the node kernel
//  - per-layer node kernel fuses: h = relu(agg*din + b_prev); x = h*dout;
//    hw = x @ W  via V_WMMA_F32_16X16X4_F32 (exact fp32, K=20 = 5 chunks of 4)
//  - per-layer edge kernel: agg[dst] += hw[src]  via global_atomic_add_f32
//  - all row buffers padded to NP=ceil64(N) rows so store epilogues need no
//    per-row bounds checks (padding rows are never read)
// ---------------------------------------------------------------------------

typedef float v2f __attribute__((ext_vector_type(2)));
typedef float v8f __attribute__((ext_vector_type(8)));

#define DD 20   // hidden width

__device__ __forceinline__ void atomAddF(float* p, float v) {
  unsafeAtomicAdd(p, v);   // hardware global_atomic_add_f32, no return
}

// --- degree kernels ---------------------------------------------------------
__global__ __launch_bounds__(256) void gcn_deg_init(float* outdeg, float* indeg, int N) {
  int n = blockIdx.x * 256 + threadIdx.x;
  if (n < N) { outdeg[n] = 1.0f; indeg[n] = 1.0f; }   // self-loop contributes 1
}

__global__ __launch_bounds__(256) void gcn_deg_count(const int* __restrict__ src,
                                                     const int* __restrict__ dst,
                                                     float* outdeg, float* indeg, int E) {
  int e = blockIdx.x * 256 + threadIdx.x;
  if (e < E) {
    atomAddF(&outdeg[src[e]], 1.0f);
    atomAddF(&indeg[dst[e]], 1.0f);
  }
}

__global__ __launch_bounds__(256) void gcn_deg_finalize(float* outdeg, float* indeg, int N) {
  int n = blockIdx.x * 256 + threadIdx.x;
  if (n < N) {                       // deg >= 1 always (self-loop)
    outdeg[n] = 1.0f / sqrtf(outdeg[n]);   // becomes dout
    indeg[n]  = 1.0f / sqrtf(indeg[n]);    // becomes din
  }
}

// --- layer 0: D_in = 1, pure rank-1 product --------------------------------
__global__ __launch_bounds__(256) void gcn_node0(const float* __restrict__ feat,
                                                 const float* __restrict__ dout,
                                                 const float* __restrict__ Ws,
                                                 float* __restrict__ hw_out,
                                                 float* __restrict__ agg_out, int N) {
  int n = blockIdx.x * 256 + threadIdx.x;
  if (n >= N) return;
  float v = feat[n] * dout[n];
  float* hp = hw_out  + (size_t)n * DD;
  float* ap = agg_out + (size_t)n * DD;
#pragma unroll
  for (int d = 0; d < DD; ++d) {
    float x = v * Ws[d];
    hp[d] = x;
    ap[d] = x;   // self-loop init
  }
}

// --- fused node kernel: finalize prev layer + GEMM via fp32 WMMA ------------
// One wave = 16 nodes. 8 waves / block = 128 nodes / block.
__global__ __launch_bounds__(256) void gcn_node_wmma(
    const float* __restrict__ agg_in,   // [NP][20] pre-norm aggregate (prev layer)
    const float* __restrict__ din,
    const float* __restrict__ dout,
    const float* __restrict__ b_prev,   // [20] bias of prev layer
    const float* __restrict__ W,        // [20][20] weight of this layer, row-major K x Ncol
    float* __restrict__ hw_out,         // [NP][20]
    float* __restrict__ agg_out,        // [NP][20] (self-loop init)
    int N) {
  const int lane = threadIdx.x & 31;
  const int wave = threadIdx.x >> 5;
  const int nb   = (blockIdx.x * 8 + wave) * 16;
  if (nb >= N) return;                 // wave-uniform exit; EXEC stays all-ones

  const int half = lane >> 4;          // 0: lanes 0-15, 1: lanes 16-31
  const int l16  = lane & 15;

  // ---- A fragments (16x4 f32 per chunk): lane holds row nb+l16 ------------
  // ISA layout: lanes 0-15 -> K = {4c+0, 4c+1}; lanes 16-31 -> K = {4c+2, 4c+3}
  int arow = nb + l16;
  if (arow >= N) arow = N - 1;         // clamp loads (N % 16 == 0 in practice)
  const float dn = din[arow];
  const float dg = dout[arow];
  const float* rowp = agg_in + (size_t)arow * DD;
  v2f a[5];
#pragma unroll
  for (int c = 0; c < 5; ++c) {
    const int k0 = 4 * c + half * 2;
    float x0 = fmaxf(rowp[k0]     * dn + b_prev[k0],     0.0f);   // relu(agg*din+b)
    float x1 = fmaxf(rowp[k0 + 1] * dn + b_prev[k0 + 1], 0.0f);
    a[c].x = x0 * dg;                                             // * dout
    a[c].y = x1 * dg;
  }

#pragma unroll
  for (int t = 0; t < 2; ++t) {        // two 16-col output tiles (cols 16..31 padded)
    const int wcol = t * 16 + l16;
    const bool cv = (wcol < DD);       // compile-time true for t=0 after unroll
    const int wc  = cv ? wcol : (DD - 1);   // clamped: loads always in-bounds
    // ---- B fragments (4x16 f32 per chunk), rows striped across lane halves:
    //      VGPR0 -> K = 4c + half, VGPR1 -> K = 4c + 2 + half   (assumed layout)
    // Unconditional (speculatable) loads + select-to-zero: no exec branches.
    v2f bf[5];
#pragma unroll
    for (int c = 0; c < 5; ++c) {
      const int kr = 4 * c + half;
      const float bx = W[(size_t)kr * DD + wc];
      const float by = W[(size_t)(kr + 2) * DD + wc];
      bf[c].x = cv ? bx : 0.0f;
      bf[c].y = cv ? by : 0.0f;
    }

    v8f acc = {};
#if defined(__HIP_DEVICE_COMPILE__)
#pragma unroll
    for (int c = 0; c < 5; ++c) {
      acc = __builtin_amdgcn_wmma_f32_16x16x4_f32(
          /*neg_a=*/false, a[c], /*neg_b=*/false, bf[c],
          /*c_mod=*/(short)0, acc, /*reuse_a=*/false, /*reuse_b=*/false);
    }
#else
    // host-pass placeholder only (never executed); keeps host sema trivial
    acc.s0 += a[0].x * bf[0].x;
#endif

    // ---- D layout: VGPR i -> row nb + i + half*8, col = l16 (+16 for t=1)
    // Rows nb..nb+15 < NP always (buffers padded); padding rows never read,
    // so no per-row bounds check. Single loop-invariant column guard.
    const size_t rbase = (size_t)(nb + half * 8) * DD + wcol;
    float* hp = hw_out  + rbase;
    float* ap = agg_out + rbase;
    if (cv) {
#pragma unroll
      for (int i = 0; i < 8; ++i) {
        hp[i * DD] = acc[i];           // immediate-offset b32 stores, 64B/half-wave
        ap[i * DD] = acc[i];           // self-loop init
      }
    }
  }
}

// --- edge scatter: agg[dst] += hw[src] --------------------------------------
__global__ __launch_bounds__(256) void gcn_scatter(const int* __restrict__ src,
                                                   const int* __restrict__ dst,
                                                   const float* __restrict__ hw,
                                                   float* __restrict__ agg, int E) {
  int e = blockIdx.x * 256 + threadIdx.x;
  if (e >= E) return;
  const int s = src[e];
  const int t = dst[e];
  const float4* sp = (const float4*)(hw + (size_t)s * DD);   // 80B rows, 16B aligned
  float* tp = agg + (size_t)t * DD;
#pragma unroll
  for (int c = 0; c < 5; ++c) {
    float4 v = sp[c];
    atomAddF(tp + 4 * c + 0, v.x);
    atomAddF(tp + 4 * c + 1, v.y);
    atomAddF(tp + 4 * c + 2, v.z);
    atomAddF(tp + 4 * c + 3, v.w);
  }
}

// --- output epilogue: out = agg*din + b_final (no relu) ---------------------
__global__ __launch_bounds__(256) void gcn_output(const float* __restrict__ agg,
                                                  const float* __restrict__ din,
                                                  const float* __restrict__ b_final,
                                                  float* __restrict__ out, int N) {
  int idx = blockIdx.x * 256 + threadIdx.x;
  if (idx >= N * DD) return;
  const int n = idx / DD;
  const int d = idx - n * DD;
  out[idx] = agg[idx] * din[n] + b_final[d];
}

// ---------------------------------------------------------------------------
extern "C" void kernel_launch(void* const* d_in, const int* in_sizes, int n_in,
                              void* d_out, int out_size, void* d_ws, size_t ws_size,
                              hipStream_t stream) {
  const float* feat    = (const float*)d_in[0];
  const float* W_start = (const float*)d_in[1];
  const float* b_start = (const float*)d_in[2];
  const float* W_mid   = (const float*)d_in[3];
  const float* b_mid   = (const float*)d_in[4];
  const float* W_final = (const float*)d_in[5];
  const float* b_final = (const float*)d_in[6];
  const int*   src     = (const int*)d_in[7];
  const int*   dst     = (const int*)d_in[8];
  float* out = (float*)d_out;

  const int N    = in_sizes[0];                // 100000
  const int E    = in_sizes[7];                // 3200000
  const int NMID = in_sizes[3] / (DD * DD);    // 18

  // workspace layout (floats), each region padded to 64-row alignment
  const size_t NP = ((size_t)N + 63) & ~(size_t)63;
  float* ws    = (float*)d_ws;
  float* doutv = ws;                 // N  : D_out^{-1/2}
  float* dinv  = ws + NP;            // N  : D_in^{-1/2}
  float* hw    = ws + 2 * NP;        // NP*20 : gather source
  float* aggA  = hw  + DD * NP;      // NP*20
  float* aggB  = aggA + DD * NP;     // NP*20

  const int gN = (N + 255) / 256;
  const int gE = (E + 255) / 256;
  const int gW = (N + 127) / 128;            // 8 waves * 16 nodes per block
  const int gO = (N * DD + 255) / 256;

  gcn_deg_init    <<<gN, 256, 0, stream>>>(doutv, dinv, N);
  gcn_deg_count   <<<gE, 256, 0, stream>>>(src, dst, doutv, dinv, E);
  gcn_deg_finalize<<<gN, 256, 0, stream>>>(doutv, dinv, N);

  // layer 0 (D_in = 1)
  gcn_node0  <<<gN, 256, 0, stream>>>(feat, doutv, W_start, hw, aggA, N);
  gcn_scatter<<<gE, 256, 0, stream>>>(src, dst, hw, aggA, E);

  // layers 1..18 (W_mid) and layer 19 (W_final); all finalize with relu
  float* cur = aggA;
  float* nxt = aggB;
  for (int i = 0; i <= NMID; ++i) {
    const float* W  = (i < NMID) ? (W_mid + (size_t)i * DD * DD) : W_final;
    const float* bp = (i == 0) ? b_start : (b_mid + (size_t)(i - 1) * DD);
    gcn_node_wmma<<<gW, 256, 0, stream>>>(cur, dinv, doutv, bp, W, hw, nxt, N);
    gcn_scatter  <<<gE, 256, 0, stream>>>(src, dst, hw, nxt, E);
    float* tmp = cur; cur = nxt; nxt = tmp;
  }

  // final: out = agg * din + b_final   (no relu)
  gcn_output<<<gO, 256, 0, stream>>>(cur, dinv, b_final, out, N);
}